// CannyEdgeDetector_57269093924885
// MI455X (gfx1250) — compile-verified
//
#include <hip/hip_runtime.h>
#include <hip/hip_bf16.h>

// ---------------------------------------------------------------------------
// Canny edge detector, fused 3-kernel pipeline for MI455X (gfx1250).
// Memory-bound (~352MB @ 23.3TB/s => ~15us floor). Uses:
//   - global_load_async_to_lds_b64 + s_wait_asynccnt (CDNA5 async staging)
//   - v_wmma_f32_16x16x4_f32 for the horizontal Gaussian (banded-matrix GEMM)
// ---------------------------------------------------------------------------

typedef __attribute__((ext_vector_type(2))) float v2f;
typedef __attribute__((ext_vector_type(8))) float v8f;

#define GW0 0.05448868454964295f
#define GW1 0.24420134200863558f
#define GW2 0.40261994688344290f
#define THRESH 0.2f

// banded Gaussian matrix entry: g[d] for d in [0,4], else 0
__device__ __forceinline__ float bandw(int d) {
  float r = 0.f;
  r = (d == 0 || d == 4) ? GW0 : r;
  r = (d == 1 || d == 3) ? GW1 : r;
  r = (d == 2) ? GW2 : r;
  return r;
}

// ---------------------------------------------------------------------------
// Kernel 1: separable 5x5 Gaussian blur. One wave (32 threads) per 16x16 tile.
//   stage:  sIn[20 rows][20 cols] (pitch 20, input cols x0-2..x0+17, rows y0-2..y0+17)
//   WMMA:   H = I * G over two 16-row blocks covering H rows y0-2..y0+17
//   VALU:   vertical 5-tap from sH, write blurred tile
// ---------------------------------------------------------------------------
__global__ __launch_bounds__(32) void canny_k1_blur(const float* __restrict__ img,
                                                    float* __restrict__ outBlur,
                                                    int W, int Hh) {
  __shared__ float sIn[400]; // 20x20, pitch 20
  __shared__ float sH[400];  // 20 H-rows x 16 cols, pitch 20 (conflict-free)

  const int l  = threadIdx.x;
  const int x0 = blockIdx.x << 4;
  const int y0 = blockIdx.y << 4;
  const int p  = blockIdx.z;
  const float* gsrc = img + ((size_t)p * W * Hh);

  const bool border = (x0 == 0) || (y0 == 0) || (x0 + 16 == W) || (y0 + 16 == Hh);
  if (border) {
    for (int i = l; i < 400; i += 32) sIn[i] = 0.f;
    // make sure zeros have landed in LDS before async writes can arrive
    asm volatile("s_wait_dscnt 0" ::: "memory");
  }

  // low 32 bits of the generic shared pointer == wave-relative LDS byte offset
  unsigned ldsIn = (unsigned)(unsigned long long)(void*)sIn;

  // 20 rows x 10 b64 segments (8B each, 8B-aligned since x0 is 16-aligned)
  for (int seg = l; seg < 200; seg += 32) {
    int row = seg / 10;
    int c   = seg - row * 10;
    int gy  = y0 - 2 + row;
    int gx  = x0 - 2 + (c << 1);
    if (gy >= 0 && gy < Hh && gx >= 0 && gx < W) {  // segments never straddle edges
      unsigned la = ldsIn + (unsigned)(((row * 20 + (c << 1)) << 2));
      const float* ga = gsrc + (size_t)gy * W + gx;
      asm volatile("global_load_async_to_lds_b64 %0, %1, off" :: "v"(la), "v"(ga) : "memory");
    }
  }
  asm volatile("s_wait_asynccnt 0" ::: "memory");

  const int half = l >> 4;   // lane half selects K pair (A) / K pair (B)
  const int n    = l & 15;   // A row index == B/D column index

  // two 16-row WMMA blocks: sIn row bases 0 and 4 cover H rows -2..17
  for (int blk = 0; blk < 2; ++blk) {
    const int rb = blk ? 4 : 0;
    v8f acc = {};
    #pragma unroll
    for (int k0 = 0; k0 < 20; k0 += 4) {
      // A fragment (16x4 f32): V0 = A[m][k0 + 2*half], V1 = A[m][k0 + 2*half + 1]
      v2f a, b;
      const int row = rb + n;
      a.x = sIn[row * 20 + k0 + half * 2 + 0];
      a.y = sIn[row * 20 + k0 + half * 2 + 1];
      // B fragment (4x16 f32, banded Gaussian): V0 = B[k0+2*half][n], V1 = B[k0+2*half+1][n]
      const int kk = k0 + half * 2;
      b.x = bandw(kk - n);
      b.y = bandw(kk + 1 - n);
      acc = __builtin_amdgcn_wmma_f32_16x16x4_f32(
          /*neg_a=*/false, a, /*neg_b=*/false, b,
          /*c_mod=*/(short)0, acc, /*reuse_a=*/false, /*reuse_b=*/false);
    }
    // D layout: VGPR v = row v (lanes 0-15) / row v+8 (lanes 16-31), col = lane%16
    #pragma unroll
    for (int v = 0; v < 8; ++v) {
      int hrow = rb + v + half * 8;     // sH row index = H row + 2
      sH[hrow * 20 + n] = acc[v];
    }
  }

  // vertical 5-tap; lane handles col n, rows half*8 .. half*8+7
  float* outp = outBlur + ((size_t)p * W * Hh);
  const int rbase = half * 8;
  #pragma unroll
  for (int r = 0; r < 8; ++r) {
    const int y = rbase + r;
    const float* hp = &sH[y * 20 + n];  // sH idx y+t == H row (y + t - 2)
    float v = GW0 * hp[0] + GW1 * hp[20] + GW2 * hp[40] + GW1 * hp[60] + GW0 * hp[80];
    outp[(size_t)(y0 + y) * W + (x0 + n)] = v;
  }
}

// ---------------------------------------------------------------------------
// Kernel 2: Sobel (channel mean) + magnitude + quantized orientation + threshold
// 256 threads per 32x32 tile; stage blurred 34x34 x 3 channels with zero halo.
// ---------------------------------------------------------------------------
__global__ __launch_bounds__(256) void canny_k2_grad(const float* __restrict__ blur,
                                                     float* __restrict__ mag,
                                                     float* __restrict__ ori,
                                                     float* __restrict__ tmag,
                                                     int W, int Hh) {
  __shared__ float s[3 * 34 * 34];
  const int bx = blockIdx.x << 5, by = blockIdx.y << 5, b = blockIdx.z;
  const int tid = threadIdx.x;
  const size_t plane = (size_t)W * Hh;
  const float* base = blur + (size_t)b * 3 * plane;

  for (int i = tid; i < 3 * 1156; i += 256) {
    int ch = i / 1156, r = i - ch * 1156;
    int iy = r / 34, ix = r - iy * 34;
    int gy = by - 1 + iy, gx = bx - 1 + ix;
    float v = 0.f;
    if (gy >= 0 && gy < Hh && gx >= 0 && gx < W)
      v = base[(size_t)ch * plane + (size_t)gy * W + gx];
    s[i] = v;
  }
  __syncthreads();

  #pragma unroll
  for (int i = 0; i < 4; ++i) {
    const int px = tid + (i << 8);
    const int y = px >> 5, x = px & 31;
    float gxs = 0.f, gys = 0.f;
    #pragma unroll
    for (int ch = 0; ch < 3; ++ch) {
      const float* p0 = &s[ch * 1156 + y * 34 + x];
      float a00 = p0[0],  a01 = p0[1],  a02 = p0[2];
      float a10 = p0[34],               a12 = p0[36];
      float a20 = p0[68], a21 = p0[69], a22 = p0[70];
      gxs += (a00 - a02) + 2.f * (a10 - a12) + (a20 - a22);
      gys += (a00 + 2.f * a01 + a02) - (a20 + 2.f * a21 + a22);
    }
    gxs *= (1.f / 3.f);
    gys *= (1.f / 3.f);
    float m  = sqrtf(gxs * gxs + gys * gys);
    float od = atan2f(gys, gxs) * 57.295779513082323f + 180.f; // deg in [0,360]
    od = rintf(od * (1.f / 45.f)) * 45.f;                      // round-half-even like jnp.round
    size_t o = (size_t)b * plane + (size_t)(by + y) * W + (bx + x);
    mag[o]  = m;
    ori[o]  = od;
    tmag[o] = (m < THRESH) ? 0.f : m;
  }
}

// ---------------------------------------------------------------------------
// Kernel 3: 8-direction non-maximum suppression + threshold
// thin = (mag > neighbor_fwd && mag > neighbor_bwd along quantized axis) ? mag : 0
// Zero halo matches the reference's SAME zero-padded NMS convolution.
// ---------------------------------------------------------------------------
__global__ __launch_bounds__(256) void canny_k3_nms(const float* __restrict__ mag,
                                                    const float* __restrict__ ori,
                                                    float* __restrict__ thin,
                                                    float* __restrict__ tthin,
                                                    int W, int Hh) {
  __shared__ float s[34 * 34];
  const int bx = blockIdx.x << 5, by = blockIdx.y << 5, b = blockIdx.z;
  const int tid = threadIdx.x;
  const size_t plane = (size_t)W * Hh;
  const size_t pb = (size_t)b * plane;

  for (int i = tid; i < 1156; i += 256) {
    int iy = i / 34, ix = i - iy * 34;
    int gy = by - 1 + iy, gx = bx - 1 + ix;
    float v = 0.f;
    if (gy >= 0 && gy < Hh && gx >= 0 && gx < W) v = mag[pb + (size_t)gy * W + gx];
    s[i] = v;
  }
  __syncthreads();

  // offs (dy,dx) for dir k: packed 2-bit (val+1) tables
  // dy: {0,1,1,1,0,-1,-1,-1} -> 425 ; dx: {1,1,0,-1,-1,-1,0,1} -> 36890
  #pragma unroll
  for (int i = 0; i < 4; ++i) {
    const int px = tid + (i << 8);
    const int y = px >> 5, x = px & 31;
    size_t o = pb + (size_t)(by + y) * W + (bx + x);
    float m  = s[(y + 1) * 34 + (x + 1)];
    float od = ori[o];
    int k  = ((int)rintf(od * (1.f / 45.f))) & 7;
    int dy = (int)((425u   >> (2 * k)) & 3u) - 1;
    int dx = (int)((36890u >> (2 * k)) & 3u) - 1;
    float n1 = s[(y + 1 + dy) * 34 + (x + 1 + dx)];
    float n2 = s[(y + 1 - dy) * 34 + (x + 1 - dx)];
    float t = ((m > n1) && (m > n2)) ? m : 0.f;
    thin[o]  = t;
    tthin[o] = (t < THRESH) ? 0.f : t;
  }
}

// ---------------------------------------------------------------------------
extern "C" void kernel_launch(void* const* d_in, const int* in_sizes, int n_in,
                              void* d_out, int out_size, void* d_ws, size_t ws_size,
                              hipStream_t stream) {
  (void)d_ws; (void)ws_size; (void)n_in; (void)out_size;
  const int W = 512, H = 512, C = 3;
  const int nplanes = in_sizes[0] / (W * H);   // B*C
  const int B = nplanes / C;
  if (B <= 0) return;

  const float* img = (const float*)d_in[0];
  float* out   = (float*)d_out;
  const size_t plane = (size_t)W * H;
  float* oBlur  = out;                         // (B,3,H,W)
  float* oMag   = oBlur  + (size_t)B * C * plane;
  float* oOri   = oMag   + (size_t)B * plane;
  float* oTmag  = oOri   + (size_t)B * plane;
  float* oThin  = oTmag  + (size_t)B * plane;
  float* oTthin = oThin  + (size_t)B * plane;

  dim3 g1(W / 16, H / 16, nplanes);
  canny_k1_blur<<<g1, 32, 0, stream>>>(img, oBlur, W, H);

  dim3 g2(W / 32, H / 32, B);
  canny_k2_grad<<<g2, 256, 0, stream>>>(oBlur, oMag, oOri, oTmag, W, H);

  dim3 g3(W / 32, H / 32, B);
  canny_k3_nms<<<g3, 256, 0, stream>>>(oMag, oOri, oThin, oTthin, W, H);
}